// FAUCooccurrenceGNN_24756191494588
// MI455X (gfx1250) — compile-verified
//
#include <hip/hip_runtime.h>

#define DIM_IN  768
#define DIM_HID 512
#define DIM_OUT 256

typedef __attribute__((ext_vector_type(2))) float v2f;
typedef __attribute__((ext_vector_type(8))) float v8f;

// ---------------------------------------------------------------------------
// C[M,N] = A[M,K] @ B[K,N], f32, via V_WMMA_F32_16X16X4_F32.
// 256-thread block (8 wave32 waves) owns ROWS = 16*MT rows of C.
// Each wave computes MT x NT 16x16 tiles; every B fragment feeds MT WMMAs.
// A is staged through LDS in K-chunks of 256 (33 KB, conflict-free pad).
// B streams from global (fully L2-resident: W1 = 1.5 MB, W2 = 0.5 MB).
// Requires: 8*NT*16 == N, K % 256 == 0, gridDim.x*ROWS rows available.
// ---------------------------------------------------------------------------
template<int K, int N, int NT, int MT>
__global__ __launch_bounds__(256) void gemm_f32_wmma(
    const float* __restrict__ A, const float* __restrict__ B,
    float* __restrict__ C) {
  constexpr int KC   = 256;                 // K-chunk staged in LDS
  constexpr int KP   = KC + 4;              // padded row stride (260%64==4)
  constexpr int ROWS = 16 * MT;
  constexpr int TPR  = 256 / ROWS;          // loader threads per row
  __shared__ float As[ROWS * KP];

  const int bm = blockIdx.x;
  const float* Ab = A + (size_t)bm * ROWS * K;

  const int lane = threadIdx.x & 31;
  const int wave = threadIdx.x >> 5;
  const int r    = lane & 15;               // A row / B,C column-in-tile
  const int koff = (lane >> 4) * 2;         // K sub-offset per half-wave
  const int n0   = wave * (NT * 16);

  const int lrow = threadIdx.x / TPR;       // loader: row in tile
  const int lci  = threadIdx.x % TPR;       // loader: float4 slot

  v8f acc[MT][NT] = {};

  for (int kc = 0; kc < K; kc += KC) {
    __syncthreads();                        // previous chunk fully consumed
    // stage ROWS x KC chunk of A (float4, coalesced)
    for (int c = lci; c < KC / 4; c += TPR) {
      const float4 v = *(const float4*)&Ab[(size_t)lrow * K + kc + c * 4];
      *(float4*)&As[lrow * KP + c * 4] = v;
    }
    __syncthreads();

    for (int k0 = 0; k0 < KC; k0 += 4) {
      // A 16x4 frags: lanes 0-15 K={k0,k0+1}, lanes 16-31 K={k0+2,k0+3}
      v2f a[MT];
      #pragma unroll
      for (int m = 0; m < MT; ++m)
        a[m] = *(const v2f*)&As[(m * 16 + r) * KP + k0 + koff];

      const int kA = kc + k0 + koff;        // absolute K row of B
      #pragma unroll
      for (int t = 0; t < NT; ++t) {
        const int n = n0 + t * 16 + r;
        v2f b;                              // B 4x16 frag (mirror of A layout)
        b.x = B[(size_t)(kA)     * N + n];
        b.y = B[(size_t)(kA + 1) * N + n];
        #pragma unroll
        for (int m = 0; m < MT; ++m)
          acc[m][t] = __builtin_amdgcn_wmma_f32_16x16x4_f32(
              false, a[m], false, b, (short)0, acc[m][t], false, false);
      }
    }
  }

  // C/D layout: VGPR v -> M = v (lanes 0-15) / 8+v (lanes 16-31); N = lane&15
  const int mb = (lane >> 4) * 8;
  #pragma unroll
  for (int m = 0; m < MT; ++m) {
    #pragma unroll
    for (int t = 0; t < NT; ++t) {
      const int n = n0 + t * 16 + r;
      #pragma unroll
      for (int v = 0; v < 8; ++v)
        C[((size_t)bm * ROWS + m * 16 + mb + v) * N + n] = acc[m][t][v];
    }
  }
}

// ---------------------------------------------------------------------------
// Degree / normalization helpers
// ---------------------------------------------------------------------------
__global__ void fill_ones_kernel(float* __restrict__ p, int n) {
  int i = blockIdx.x * blockDim.x + threadIdx.x;
  if (i < n) p[i] = 1.0f;                   // self-loop contributes 1
}

__global__ void count_deg_kernel(const long long* __restrict__ dst,
                                 float* __restrict__ deg, int ne) {
  int i = blockIdx.x * blockDim.x + threadIdx.x;
  if (i < ne) unsafeAtomicAdd(&deg[(int)dst[i]], 1.0f);
}

__global__ void rsqrt_inplace_kernel(float* __restrict__ p, int n) {
  int i = blockIdx.x * blockDim.x + threadIdx.x;
  if (i < n) p[i] = rsqrtf(p[i]);           // deg >= 1 always (self-loops)
}

// Out[i][f] = b[f] + H[i][f] * dinv[i]^2   (bias + self-loop term, fused init)
template<int F>
__global__ void self_bias_kernel(const float* __restrict__ H,
                                 const float* __restrict__ b,
                                 const float* __restrict__ dinv,
                                 float* __restrict__ Out, int nn) {
  constexpr int F4 = F / 4;
  size_t idx = (size_t)blockIdx.x * blockDim.x + threadIdx.x;
  if (idx >= (size_t)nn * F4) return;
  const int node = (int)(idx / F4);
  const int f    = (int)(idx % F4) * 4;
  const float di = dinv[node];
  const float w  = di * di;
  const float4 h = *(const float4*)(H + (size_t)node * F + f);
  float4 o;
  o.x = b[f + 0] + h.x * w;
  o.y = b[f + 1] + h.y * w;
  o.z = b[f + 2] + h.z * w;
  o.w = b[f + 3] + h.w * w;
  *(float4*)(Out + (size_t)node * F + f) = o;
}

// Out[dst] += H[src] * dinv[src]*dinv[dst]  over all edges (F/4 threads/edge)
template<int F>
__global__ void edge_scatter_kernel(const float* __restrict__ H,
                                    const float* __restrict__ dinv,
                                    const long long* __restrict__ src,
                                    const long long* __restrict__ dst,
                                    float* __restrict__ Out, int ne) {
  constexpr int TPE = F / 4;
  size_t gid = (size_t)blockIdx.x * blockDim.x + threadIdx.x;
  size_t e = gid / TPE;
  if (e >= (size_t)ne) return;
  const int f = (int)(gid % TPE) * 4;
  const int s = (int)src[e];
  const int d = (int)dst[e];
  const float w = dinv[s] * dinv[d];
  const float4 h = *(const float4*)(H + (size_t)s * F + f);
  float* o = Out + (size_t)d * F + f;
  unsafeAtomicAdd(o + 0, h.x * w);
  unsafeAtomicAdd(o + 1, h.y * w);
  unsafeAtomicAdd(o + 2, h.z * w);
  unsafeAtomicAdd(o + 3, h.w * w);
}

__global__ void relu_inplace_kernel(float* __restrict__ p, size_t n4) {
  size_t i = (size_t)blockIdx.x * blockDim.x + threadIdx.x;
  if (i >= n4) return;
  float4 v = ((float4*)p)[i];
  v.x = fmaxf(v.x, 0.0f);
  v.y = fmaxf(v.y, 0.0f);
  v.z = fmaxf(v.z, 0.0f);
  v.w = fmaxf(v.w, 0.0f);
  ((float4*)p)[i] = v;
}

// ---------------------------------------------------------------------------
template<int K, int N, int NT>
static void launch_gemm(const float* A, const float* B, float* C, int nn,
                        hipStream_t stream) {
  const int nb2 = nn / 32;                  // MT=2 blocks (32 rows each)
  if (nb2 > 0)
    gemm_f32_wmma<K, N, NT, 2><<<nb2, 256, 0, stream>>>(A, B, C);
  const int rem = nn - nb2 * 32;            // leftover rows (multiple of 16)
  if (rem > 0) {
    const size_t moff = (size_t)nb2 * 32;
    gemm_f32_wmma<K, N, NT, 1><<<rem / 16, 256, 0, stream>>>(
        A + moff * K, B, C + moff * N);
  }
}

extern "C" void kernel_launch(void* const* d_in, const int* in_sizes, int n_in,
                              void* d_out, int out_size, void* d_ws, size_t ws_size,
                              hipStream_t stream) {
  const float*     x   = (const float*)d_in[0];
  const long long* ei  = (const long long*)d_in[1];   // int64 [2, E]
  const float*     W1  = (const float*)d_in[2];
  const float*     b1  = (const float*)d_in[3];
  const float*     W2  = (const float*)d_in[4];
  const float*     b2  = (const float*)d_in[5];
  float* out = (float*)d_out;

  const int nn = in_sizes[0] / DIM_IN;     // 50000
  const int ne = in_sizes[1] / 2;          // 800000
  const long long* src = ei;
  const long long* dst = ei + ne;

  // Workspace layout (bytes):
  //   [0, nn*4)                    : dinv
  //   [1MB, 1MB + nn*512*4)        : H1 (layer-2 H2 reuses this buffer)
  //   [1MB + nn*512*4, +nn*512*4)  : A1 (relu'd hidden activations)
  char* ws = (char*)d_ws;
  float* dinv = (float*)ws;
  size_t hbytes = (size_t)nn * DIM_HID * sizeof(float);   // 102,400,000
  float* H1 = (float*)(ws + (1u << 20));
  float* A1 = (float*)(ws + (1u << 20) + hbytes);
  float* H2 = H1;                                         // reuse after A1 built

  const int T = 256;

  // 1) degrees (incl. self-loop) -> dinv = 1/sqrt(deg)
  fill_ones_kernel<<<(nn + T - 1) / T, T, 0, stream>>>(dinv, nn);
  count_deg_kernel<<<(ne + T - 1) / T, T, 0, stream>>>(dst, dinv, ne);
  rsqrt_inplace_kernel<<<(nn + T - 1) / T, T, 0, stream>>>(dinv, nn);

  // 2) H1 = X @ W1   (50000x768 @ 768x512)
  launch_gemm<DIM_IN, DIM_HID, 4>(x, W1, H1, nn, stream);

  // 3) A1 = b1 + self-loop + edge scatter; then ReLU
  {
    size_t n4 = (size_t)nn * (DIM_HID / 4);
    self_bias_kernel<DIM_HID><<<(unsigned)((n4 + T - 1) / T), T, 0, stream>>>(
        H1, b1, dinv, A1, nn);
    size_t tot = (size_t)ne * (DIM_HID / 4);
    edge_scatter_kernel<DIM_HID><<<(unsigned)((tot + T - 1) / T), T, 0, stream>>>(
        H1, dinv, src, dst, A1, ne);
    relu_inplace_kernel<<<(unsigned)((n4 + T - 1) / T), T, 0, stream>>>(A1, n4);
  }

  // 4) H2 = A1 @ W2   (50000x512 @ 512x256)
  launch_gemm<DIM_HID, DIM_OUT, 2>(A1, W2, H2, nn, stream);

  // 5) out = b2 + self-loop + edge scatter
  {
    size_t n4 = (size_t)nn * (DIM_OUT / 4);
    self_bias_kernel<DIM_OUT><<<(unsigned)((n4 + T - 1) / T), T, 0, stream>>>(
        H2, b2, dinv, out, nn);
    size_t tot = (size_t)ne * (DIM_OUT / 4);
    edge_scatter_kernel<DIM_OUT><<<(unsigned)((tot + T - 1) / T), T, 0, stream>>>(
        H2, dinv, src, dst, out, ne);
  }
}